// NeuroKernel_56590489092176
// MI455X (gfx1250) — compile-verified
//
#include <hip/hip_runtime.h>

// ---------------------------------------------------------------------------
// N=2048. Phase 1: K[i][j] = MLP(x[i],x[j]) for j>=i else 0; the dominant
// 128->32 layer runs on the matrix pipe (V_WMMA_F32_16X16X4_F32, full f32).
// Phase 2: out = K^T @ K, LDS-tiled f32 WMMA GEMM with:
//   - triangular k-cutoff (K upper-triangular => kend = min(m0+BM, n0+BN)),
//   - Tensor Data Mover (tensor_load_to_lds) double-buffered staging,
//     tracked by TENSORcnt, one barrier per k-chunk.
// Compute-bound: ~50MB total HBM traffic vs 23.3 TB/s.
// ---------------------------------------------------------------------------

#define NDIM 2048

typedef float v2f __attribute__((ext_vector_type(2)));
typedef float v8f __attribute__((ext_vector_type(8)));
typedef unsigned int u32;
typedef u32 u32x4 __attribute__((ext_vector_type(4)));
typedef u32 u32x8 __attribute__((ext_vector_type(8)));

// LDS byte offset of a __shared__ object (generic -> AS(3) -> int).
__device__ __forceinline__ u32 lds_offset(const void* p) {
    return (u32)(unsigned long long)
        (__attribute__((address_space(3))) const unsigned char*)p;
}

// Issue one TDM 2D tile load: tile_h rows x tile_w f32 elements from a
// row-major f32 tensor with row stride `stride_elems`, into LDS at lds_addr
// (packed, no padding). D# layout per CDNA5 ISA ch.8. Wave-level op,
// tracked by TENSORcnt.
__device__ __forceinline__ void tdm_load_2d(u32 lds_addr, const void* gptr,
                                            u32 tile_w, u32 tile_h,
                                            u32 stride_elems) {
    const unsigned long long ga = (unsigned long long)gptr;
    u32x4 g0;
    g0[0] = 1u;                                   // count=1 (valid), user D#
    g0[1] = lds_addr;                             // LDS byte address
    g0[2] = (u32)ga;                              // global_addr[31:0]
    g0[3] = (u32)((ga >> 32) & 0x01FFFFFFu)       // global_addr[56:32]
            | (2u << 30);                         // type=2 ("image")

    u32x8 g1;
    g1[0] = 2u << 16;                 // workgroup_mask=0, data_size=2 (4B)
    g1[1] = (u32)(NDIM & 0xFFFF) << 16;           // tensor_dim0 lo16 @bit48
    g1[2] = (u32)(NDIM >> 16) | ((u32)(NDIM & 0xFFFF) << 16); // d0 hi|d1 lo
    g1[3] = (u32)(NDIM >> 16) | (tile_w << 16);   // d1 hi | tile_dim0
    g1[4] = tile_h;                               // tile_dim1 (tile_dim2=0)
    g1[5] = stride_elems;                         // tensor_dim0_stride lo32
    g1[6] = 0u;                                   // stride0 hi | stride1 lo
    g1[7] = 0u;                                   // stride1 hi
    const u32x4 gz = {0u, 0u, 0u, 0u};            // groups 2/3: zero (2D)
    asm volatile("tensor_load_to_lds %0, %1, %2, %3"
                 :: "s"(g0), "s"(g1), "s"(gz), "s"(gz)
                 : "memory");
}

// ---------------------------------------------------------------------------
// Phase 1: 128 threads = 4 waves; wave w handles pairs (i, j0+16w..j0+16w+15).
// ---------------------------------------------------------------------------
__global__ __launch_bounds__(128) void build_K_wmma_kernel(
    const float* __restrict__ x,
    const float* __restrict__ W1,  // [128,2]
    const float* __restrict__ B1,  // [128]
    const float* __restrict__ W2,  // [32,128]
    const float* __restrict__ B2,  // [32]
    const float* __restrict__ W3,  // [1,32]
    const float* __restrict__ B3,  // [1]
    float* __restrict__ K)
{
    const int i   = blockIdx.y;
    const int j0  = blockIdx.x * 64;
    const int tid = threadIdx.x;

    if (j0 + 63 < i) {               // tile strictly below diagonal -> zeros
        if (tid < 64) K[(size_t)i * NDIM + j0 + tid] = 0.0f;
        return;
    }

    __shared__ float Wt[128 * 33];   // Wt[k*33+n] = W2[n][k]
    __shared__ float Af[128];
    __shared__ float Bf[128];
    __shared__ float Cf[128];
    __shared__ float Hs[4][128 * 16];// per-wave h1, k-major: Hs[w][k*16+p]

    for (int idx = tid; idx < 4096; idx += 128) {
        int k = idx & 127;
        int n = idx >> 7;
        Wt[k * 33 + n] = W2[idx];    // idx == n*128 + k
    }
    {
        int k = tid;
        Af[k] = W1[2 * k];
        Bf[k] = W1[2 * k + 1];
        Cf[k] = B1[k];
    }
    __syncthreads();

    const int lane = tid & 31;
    const int wid  = tid >> 5;
    const int lm   = lane & 15;
    const int half = lane >> 4;
    float* H = Hs[wid];

    const float xi = x[i];
    const float xj = x[j0 + wid * 16 + lm];
    const float L2E = 1.4426950408889634f;

#pragma unroll 8
    for (int kk = 0; kk < 64; ++kk) {
        const int k = half * 64 + kk;
        float t = __builtin_fmaf(Af[k], xi, __builtin_fmaf(Bf[k], xj, Cf[k]));
        float h = __builtin_amdgcn_rcpf(1.0f + __builtin_amdgcn_exp2f(-t * L2E));
        H[k * 16 + lm] = h;
    }

    v8f acc0, acc1;
    {
        const float bn0 = B2[lm];
        const float bn1 = B2[16 + lm];
#pragma unroll
        for (int d = 0; d < 8; ++d) { acc0[d] = bn0; acc1[d] = bn1; }
    }
    const int khalf = half << 1;
#pragma unroll 8
    for (int ks = 0; ks < 32; ++ks) {
        const int kk = 4 * ks + khalf;
        v2f a  = {H[kk * 16 + lm],        H[(kk + 1) * 16 + lm]};
        v2f b0 = {Wt[kk * 33 + lm],       Wt[(kk + 1) * 33 + lm]};
        v2f b1 = {Wt[kk * 33 + 16 + lm],  Wt[(kk + 1) * 33 + 16 + lm]};
        acc0 = __builtin_amdgcn_wmma_f32_16x16x4_f32(
            false, a, false, b0, (short)0, acc0, false, false);
        acc1 = __builtin_amdgcn_wmma_f32_16x16x4_f32(
            false, a, false, b1, (short)0, acc1, false, false);
    }

    float* H2 = H;                   // overlay dead H region (same-wave LDS)
#pragma unroll
    for (int d = 0; d < 8; ++d) {
        const int p = d + 8 * half;
        H2[p * 32 + lm]      = fmaxf(acc0[d], 0.0f);
        H2[p * 32 + 16 + lm] = fmaxf(acc1[d], 0.0f);
    }

    if (lane < 16) {
        float v = B3[0];
#pragma unroll 8
        for (int n = 0; n < 32; ++n) {
            v = __builtin_fmaf(W3[n], H2[lane * 32 + n], v);
        }
        const int j = j0 + wid * 16 + lane;
        K[(size_t)i * NDIM + j] = (j >= i) ? v : 0.0f;
    }
}

// ---------------------------------------------------------------------------
// Phase 2: out = K^T @ K. TDM double-buffered staging + f32 WMMA.
// ---------------------------------------------------------------------------
#define BM 128
#define BN 64
#define KC 32

__global__ __launch_bounds__(256) void gemm_ktk_kernel(
    const float* __restrict__ K, float* __restrict__ out)
{
    __shared__ float As[2][KC][BM];  // 2 x 16 KB
    __shared__ float Bs[2][KC][BN];  // 2 x  8 KB

    const int tid  = threadIdx.x;
    const int lane = tid & 31;
    const int wid  = tid >> 5;
    const int wm   = (wid & 3) * 32;
    const int wn   = (wid >> 2) * 32;

    const int m0 = blockIdx.y * BM;
    const int n0 = blockIdx.x * BN;

    const int lm = lane & 15;
    const int kh = (lane >> 4) << 1;

    // K is upper-triangular: K[k][c]==0 for c<k, so only k < min(m0+BM, n0+BN)
    // contributes. kend is a multiple of KC (BM, BN multiples of KC).
    const int ke_a  = m0 + BM;
    const int ke_b  = n0 + BN;
    const int kend  = ke_a < ke_b ? ke_a : ke_b;

    v8f acc00 = {}, acc01 = {}, acc10 = {}, acc11 = {};

    // Prologue: DMA chunk 0 into buffer 0 (wave 0 issues; TENSORcnt tracks).
    if (wid == 0) {
        tdm_load_2d(lds_offset(&As[0][0][0]), &K[(size_t)0 * NDIM + m0],
                    BM, KC, NDIM);
        tdm_load_2d(lds_offset(&Bs[0][0][0]), &K[(size_t)0 * NDIM + n0],
                    BN, KC, NDIM);
    }

    for (int k0 = 0; k0 < kend; k0 += KC) {
        const int buf = (k0 >> 5) & 1;

        // Wave 0: wait for this chunk's DMA; others pass trivially.
        __builtin_amdgcn_s_wait_tensorcnt(0);
        // One barrier per chunk: publishes DMA'd data to all waves AND
        // guarantees every wave finished reading the other buffer before
        // the next DMA overwrites it.
        __syncthreads();

        if (wid == 0 && (k0 + KC) < kend) {
            const int kn = k0 + KC;
            tdm_load_2d(lds_offset(&As[buf ^ 1][0][0]),
                        &K[(size_t)kn * NDIM + m0], BM, KC, NDIM);
            tdm_load_2d(lds_offset(&Bs[buf ^ 1][0][0]),
                        &K[(size_t)kn * NDIM + n0], BN, KC, NDIM);
        }

#pragma unroll
        for (int ks = 0; ks < KC; ks += 4) {
            const int kk = ks + kh;
            v2f a0 = {As[buf][kk][wm + lm],      As[buf][kk + 1][wm + lm]};
            v2f a1 = {As[buf][kk][wm + 16 + lm], As[buf][kk + 1][wm + 16 + lm]};
            v2f b0 = {Bs[buf][kk][wn + lm],      Bs[buf][kk + 1][wn + lm]};
            v2f b1 = {Bs[buf][kk][wn + 16 + lm], Bs[buf][kk + 1][wn + 16 + lm]};

            acc00 = __builtin_amdgcn_wmma_f32_16x16x4_f32(
                false, a0, false, b0, (short)0, acc00, false, false);
            acc01 = __builtin_amdgcn_wmma_f32_16x16x4_f32(
                false, a0, false, b1, (short)0, acc01, false, false);
            acc10 = __builtin_amdgcn_wmma_f32_16x16x4_f32(
                false, a1, false, b0, (short)0, acc10, false, false);
            acc11 = __builtin_amdgcn_wmma_f32_16x16x4_f32(
                false, a1, false, b1, (short)0, acc11, false, false);
        }
    }

    const int mrow = (lane >> 4) << 3;
    const int gm0  = m0 + wm + mrow;
    const int gn0  = n0 + wn + lm;
#pragma unroll
    for (int d = 0; d < 8; ++d) {
        out[(size_t)(gm0 + d) * NDIM + gn0]           = acc00[d];
        out[(size_t)(gm0 + d) * NDIM + gn0 + 16]      = acc01[d];
        out[(size_t)(gm0 + 16 + d) * NDIM + gn0]      = acc10[d];
        out[(size_t)(gm0 + 16 + d) * NDIM + gn0 + 16] = acc11[d];
    }
}

// ---------------------------------------------------------------------------
extern "C" void kernel_launch(void* const* d_in, const int* in_sizes, int n_in,
                              void* d_out, int out_size, void* d_ws, size_t ws_size,
                              hipStream_t stream) {
    const float* x  = (const float*)d_in[0];
    const float* W1 = (const float*)d_in[1];
    const float* B1 = (const float*)d_in[2];
    const float* W2 = (const float*)d_in[3];
    const float* B2 = (const float*)d_in[4];
    const float* W3 = (const float*)d_in[5];
    const float* B3 = (const float*)d_in[6];

    float* K   = (float*)d_ws;           // 16 MB scratch
    float* out = (float*)d_out;

    dim3 g1(NDIM / 64, NDIM);            // (32, 2048)
    build_K_wmma_kernel<<<g1, 128, 0, stream>>>(x, W1, B1, W2, B2, W3, B3, K);

    dim3 g2(NDIM / BN, NDIM / BM);       // (32, 16)
    gemm_ktk_kernel<<<g2, 256, 0, stream>>>(K, out);
}